// GATModel_51797305590002
// MI455X (gfx1250) — compile-verified
//
#include <hip/hip_runtime.h>
#include <hip/hip_bf16.h>

#define GN 100000
#define GE 1600000
#define GEP (GE + GN)      // edges + self loops = 1,700,000
#define GH 4
#define GC 64
#define GHC 256
#define NEG_SLOPE 0.2f

typedef float v2f __attribute__((ext_vector_type(2)));
typedef float v8f __attribute__((ext_vector_type(8)));
typedef unsigned int u32x4 __attribute__((ext_vector_type(4)));
typedef int i32x4 __attribute__((ext_vector_type(4)));
typedef int i32x8 __attribute__((ext_vector_type(8)));

#if __has_builtin(__builtin_amdgcn_tensor_load_to_lds)
#define HAVE_TDM 1
#endif

// ---------- helpers ----------
__device__ __forceinline__ float leaky(float v) {
    return v > 0.f ? v : NEG_SLOPE * v;
}
// order-preserving float -> uint encoding for atomic max
__device__ __forceinline__ unsigned f2u_ord(float f) {
    unsigned u = __float_as_uint(f);
    return (u >> 31) ? ~u : (u | 0x80000000u);
}
__device__ __forceinline__ float u2f_ord(unsigned e) {
    return (e & 0x80000000u) ? __uint_as_float(e & 0x7FFFFFFFu)
                             : __uint_as_float(~e);
}

// ---------- kernel 1: h1 = x @ W1, plus per-head attention logits ----------
__global__ __launch_bounds__(256) void k_layer1(
    const float* __restrict__ x, const float* __restrict__ W1,
    const float* __restrict__ a_s, const float* __restrict__ a_d,
    float* __restrict__ h, float* __restrict__ als, float* __restrict__ ald) {
    const int n = blockIdx.x;
    const int t = threadIdx.x;
    const float x0 = x[n * 3 + 0], x1 = x[n * 3 + 1], x2 = x[n * 3 + 2];
    const float hv = x0 * W1[t] + x1 * W1[GHC + t] + x2 * W1[2 * GHC + t];
    h[(size_t)n * GHC + t] = hv;
    __shared__ float rs[GHC];
    __shared__ float rd[GHC];
    rs[t] = hv * a_s[t];
    rd[t] = hv * a_d[t];
    __syncthreads();
    // reduce within each group of 64 threads (one head)
    for (int off = 32; off >= 1; off >>= 1) {
        if ((t & 63) < off) { rs[t] += rs[t + off]; rd[t] += rd[t + off]; }
        __syncthreads();
    }
    if ((t & 63) == 0) {
        als[n * GH + (t >> 6)] = rs[t];
        ald[n * GH + (t >> 6)] = rd[t];
    }
}

// ---------- zero / init ----------
__global__ void k_zero(float* __restrict__ p, int len) {
    int i = blockIdx.x * blockDim.x + threadIdx.x;
    if (i < len) p[i] = 0.f;
}
__global__ void k_init_nh(unsigned* __restrict__ menc, float* __restrict__ den) {
    int i = blockIdx.x * blockDim.x + threadIdx.x;
    if (i < GN * GH) { menc[i] = 0x007FFFFFu; den[i] = 0.f; }  // enc(-inf)
}

// ---------- edge pass A: segment max of leaky logits ----------
__global__ void k_edge_max(const long long* __restrict__ src,
                           const long long* __restrict__ dst,
                           const float* __restrict__ als,
                           const float* __restrict__ ald,
                           unsigned* __restrict__ menc) {
    int e = blockIdx.x * blockDim.x + threadIdx.x;
    if (e >= GEP) return;
    int s, d;
    if (e < GE) { s = (int)src[e]; d = (int)dst[e]; } else { s = d = e - GE; }
    const float4 a = *(const float4*)(als + (size_t)s * GH);
    const float4 b = *(const float4*)(ald + (size_t)d * GH);
    unsigned* mp = menc + (size_t)d * GH;
    atomicMax(mp + 0, f2u_ord(leaky(a.x + b.x)));
    atomicMax(mp + 1, f2u_ord(leaky(a.y + b.y)));
    atomicMax(mp + 2, f2u_ord(leaky(a.z + b.z)));
    atomicMax(mp + 3, f2u_ord(leaky(a.w + b.w)));
}

__global__ void k_decode_max(unsigned* __restrict__ menc) {
    int i = blockIdx.x * blockDim.x + threadIdx.x;
    if (i < GN * GH) menc[i] = __float_as_uint(u2f_ord(menc[i]));
}

// ---------- edge pass B: denom = segment sum of exp(e - m) ----------
__global__ void k_edge_sum(const long long* __restrict__ src,
                           const long long* __restrict__ dst,
                           const float* __restrict__ als,
                           const float* __restrict__ ald,
                           const float* __restrict__ m,
                           float* __restrict__ den) {
    int e = blockIdx.x * blockDim.x + threadIdx.x;
    if (e >= GEP) return;
    int s, d;
    if (e < GE) { s = (int)src[e]; d = (int)dst[e]; } else { s = d = e - GE; }
    const float4 a = *(const float4*)(als + (size_t)s * GH);
    const float4 b = *(const float4*)(ald + (size_t)d * GH);
    const float4 mm = *(const float4*)(m + (size_t)d * GH);
    float* dp = den + (size_t)d * GH;
    atomicAdd(dp + 0, __expf(leaky(a.x + b.x) - mm.x));
    atomicAdd(dp + 1, __expf(leaky(a.y + b.y) - mm.y));
    atomicAdd(dp + 2, __expf(leaky(a.z + b.z) - mm.z));
    atomicAdd(dp + 3, __expf(leaky(a.w + b.w) - mm.w));
}

// ---------- edge pass C: scatter-aggregate alpha * h[src] ----------
__global__ __launch_bounds__(256) void k_edge_agg(
    const long long* __restrict__ src, const long long* __restrict__ dst,
    const float* __restrict__ als, const float* __restrict__ ald,
    const float* __restrict__ m, const float* __restrict__ den,
    const float* __restrict__ hsrc, float* __restrict__ agg) {
    long long gid = (long long)blockIdx.x * blockDim.x + threadIdx.x;
    if (gid >= (long long)GEP * 64) return;
    const int e = (int)(gid >> 6);
    const int q = (int)(gid & 63);   // float4 chunk within the 256-wide row
    const int hh = q >> 4;           // head
    int s, d;
    if (e < GE) { s = (int)src[e]; d = (int)dst[e]; } else { s = d = e - GE; }
    const float lv = leaky(als[(size_t)s * GH + hh] + ald[(size_t)d * GH + hh]);
    const float alpha = __expf(lv - m[(size_t)d * GH + hh]) /
                        (den[(size_t)d * GH + hh] + 1e-16f);
    const float4 hv = *(const float4*)(hsrc + (size_t)s * GHC + q * 4);
    float* ap = agg + (size_t)d * GHC + q * 4;
    atomicAdd(ap + 0, alpha * hv.x);
    atomicAdd(ap + 1, alpha * hv.y);
    atomicAdd(ap + 2, alpha * hv.z);
    atomicAdd(ap + 3, alpha * hv.w);
}

// ---------- bias + relu ----------
__global__ void k_post(const float* __restrict__ agg, const float* __restrict__ b,
                       float* __restrict__ o) {
    int i = blockIdx.x * blockDim.x + threadIdx.x;
    if (i < GN * GHC) o[i] = fmaxf(agg[i] + b[i & (GHC - 1)], 0.f);
}

// ---------- WMMA FP32 GEMM: O[N,256] = Z[N,256] @ W[256,256] ----------
// one wave -> one 16x16 output tile; A tile (16x256) staged in LDS by the
// Tensor Data Mover (hardware-padded to a conflict-free 260-float stride).
__global__ __launch_bounds__(256) void k_gemm_wmma(
    const float* __restrict__ Z, const float* __restrict__ W,
    float* __restrict__ O) {
    __shared__ float As[16][260];   // padded: conflict-free column reads
    const int tid = threadIdx.x;
    const int r0 = (blockIdx.x >> 1) * 16;   // 16-row block
    const int cb = blockIdx.x & 1;           // column half (8 tiles each)

#ifdef HAVE_TDM
    // --- TDM path: DMA the 16x256 f32 tile Global->LDS with LDS padding of
    //     4 DWORDs after every 256 DWORDs (=> row stride 260 floats).
    if (tid < 32) {   // single wave issues the DMA (TDM ignores EXEC)
        const unsigned long long ga =
            (unsigned long long)(const void*)(Z + (size_t)r0 * GHC);
        const unsigned ldsa = (unsigned)(size_t)(void*)&As[0][0];
        u32x4 g0;
        g0[0] = 1u;                                   // count=1, user D#
        g0[1] = ldsa;                                 // lds_addr
        g0[2] = (unsigned)ga;                         // global_addr[31:0]
        g0[3] = (unsigned)(ga >> 32) | (2u << 30);    // addr[56:32] | type=2
        i32x8 g1;
        g1[0] = (2 << 16)      // data_size = 4 bytes
              | (1 << 20)      // pad_enable
              | (7 << 22)      // pad_interval: every 256 DWORDs
              | (3 << 25);     // pad_amount: 4 DWORDs
        g1[1] = (int)(256u << 16);   // tensor_dim0 = 256 (low16)
        g1[2] = (int)(16u << 16);    // tensor_dim1 = 16  (low16)
        g1[3] = (int)(256u << 16);   // tile_dim0 = 256
        g1[4] = 16;                  // tile_dim1 = 16
        g1[5] = 256;                 // tensor_dim0_stride = 256
        g1[6] = 0;
        g1[7] = 0;
        i32x4 gz = {0, 0, 0, 0};
#if __clang_major__ >= 23
        i32x8 gz8 = {0, 0, 0, 0, 0, 0, 0, 0};
        __builtin_amdgcn_tensor_load_to_lds(g0, g1, gz, gz, gz8, 0);
#else
        __builtin_amdgcn_tensor_load_to_lds(g0, g1, gz, gz, 0);
#endif
#if __has_builtin(__builtin_amdgcn_s_wait_tensorcnt)
        __builtin_amdgcn_s_wait_tensorcnt(0);
#else
        asm volatile("s_wait_tensorcnt 0x0" ::: "memory");
#endif
    }
#else
    // --- fallback: cooperative float4 staging
    for (int p = tid; p < 1024; p += 256) {
        const int row = p >> 6;
        const int c4 = (p & 63) * 4;
        const float4 v = *(const float4*)(Z + (size_t)(r0 + row) * GHC + c4);
        As[row][c4 + 0] = v.x; As[row][c4 + 1] = v.y;
        As[row][c4 + 2] = v.z; As[row][c4 + 3] = v.w;
    }
#endif
    __syncthreads();

    const int wid = tid >> 5;
    const int lane = tid & 31;
    const int hi = lane >> 4;        // 0: K={0,1}; 1: K={2,3}
    const int mr = lane & 15;        // row (A) / col (B,C)
    const int n0 = (cb * 8 + wid) * 16;
    v8f acc = {};
#pragma unroll 4
    for (int k = 0; k < GHC; k += 4) {
        v2f a, b;
        a.x = As[mr][k + 2 * hi];
        a.y = As[mr][k + 2 * hi + 1];
        const float* wp = W + (size_t)(k + 2 * hi) * GHC + n0 + mr;
        b.x = wp[0];
        b.y = wp[GHC];
        acc = __builtin_amdgcn_wmma_f32_16x16x4_f32(
            false, a, false, b, (short)0, acc, false, false);
    }
#pragma unroll
    for (int v = 0; v < 8; ++v)
        O[(size_t)(r0 + v + 8 * hi) * GHC + n0 + mr] = acc[v];
}

// ---------- per-(node,head) logits for layer 2 ----------
__global__ void k_logits(const float* __restrict__ h,
                         const float* __restrict__ a_s,
                         const float* __restrict__ a_d,
                         float* __restrict__ als, float* __restrict__ ald) {
    int i = blockIdx.x * blockDim.x + threadIdx.x;
    if (i >= GN * GH) return;
    const int n = i >> 2, hh = i & 3;
    const float* hp = h + (size_t)n * GHC + hh * GC;
    const float* sp = a_s + hh * GC;
    const float* dp = a_d + hh * GC;
    float accs = 0.f, accd = 0.f;
#pragma unroll 4
    for (int c = 0; c < GC; ++c) { accs += hp[c] * sp[c]; accd += hp[c] * dp[c]; }
    als[i] = accs;
    ald[i] = accd;
}

// ---------- final: out = relu(agg + b2) @ fc_w + fc_b ----------
__global__ void k_final(const float* __restrict__ agg, const float* __restrict__ b2,
                        const float* __restrict__ fcw, const float* __restrict__ fcb,
                        float* __restrict__ out) {
    int n = blockIdx.x * blockDim.x + threadIdx.x;
    if (n >= GN) return;
    float a0 = fcb[0], a1 = fcb[1];
    const float* ap = agg + (size_t)n * GHC;
#pragma unroll 4
    for (int c = 0; c < GHC; ++c) {
        const float z = fmaxf(ap[c] + b2[c], 0.f);
        a0 += z * fcw[c * 2 + 0];
        a1 += z * fcw[c * 2 + 1];
    }
    out[n * 2 + 0] = a0;
    out[n * 2 + 1] = a1;
}

extern "C" void kernel_launch(void* const* d_in, const int* in_sizes, int n_in,
                              void* d_out, int out_size, void* d_ws, size_t ws_size,
                              hipStream_t stream) {
    const float* x   = (const float*)d_in[0];
    const long long* ei = (const long long*)d_in[1];   // int64 per reference
    const float* W1  = (const float*)d_in[2];
    const float* as1 = (const float*)d_in[3];
    const float* ad1 = (const float*)d_in[4];
    const float* b1  = (const float*)d_in[5];
    const float* W2  = (const float*)d_in[6];
    const float* as2 = (const float*)d_in[7];
    const float* ad2 = (const float*)d_in[8];
    const float* b2  = (const float*)d_in[9];
    const float* fcw = (const float*)d_in[10];
    const float* fcb = (const float*)d_in[11];
    float* out = (float*)d_out;
    const long long* src = ei;
    const long long* dst = ei + GE;

    // workspace layout
    float* bufA = (float*)d_ws;                       // N*256 (h1 / z1 / agg2)
    float* bufB = bufA + (size_t)GN * GHC;            // N*256 (agg1 / h2)
    float* als  = bufB + (size_t)GN * GHC;            // N*H
    float* ald  = als + (size_t)GN * GH;              // N*H
    unsigned* menc = (unsigned*)(ald + (size_t)GN * GH);  // N*H (enc max / float max)
    float* mF   = (float*)menc;
    float* den  = (float*)(menc + (size_t)GN * GH);   // N*H

    const int nhBlk = (GN * GH + 255) / 256;
    const int nodeChBlk = (GN * GHC + 255) / 256;
    const int edgeBlk = (GEP + 255) / 256;
    const long long aggWork = (long long)GEP * 64;
    const int aggBlk = (int)((aggWork + 255) / 256);

    // ---- layer 1 ----
    k_layer1<<<GN, 256, 0, stream>>>(x, W1, as1, ad1, bufA, als, ald);
    k_zero<<<nodeChBlk, 256, 0, stream>>>(bufB, GN * GHC);
    k_init_nh<<<nhBlk, 256, 0, stream>>>(menc, den);
    k_edge_max<<<edgeBlk, 256, 0, stream>>>(src, dst, als, ald, menc);
    k_decode_max<<<nhBlk, 256, 0, stream>>>(menc);
    k_edge_sum<<<edgeBlk, 256, 0, stream>>>(src, dst, als, ald, mF, den);
    k_edge_agg<<<aggBlk, 256, 0, stream>>>(src, dst, als, ald, mF, den, bufA, bufB);
    k_post<<<nodeChBlk, 256, 0, stream>>>(bufB, b1, bufA);   // z1 = relu(agg1+b1)

    // ---- layer 2 projection: h2 = z1 @ W2 (WMMA + TDM) ----
    k_gemm_wmma<<<(GN / 16) * 2, 256, 0, stream>>>(bufA, W2, bufB);
    k_logits<<<nhBlk, 256, 0, stream>>>(bufB, as2, ad2, als, ald);

    // ---- layer 2 attention ----
    k_zero<<<nodeChBlk, 256, 0, stream>>>(bufA, GN * GHC);
    k_init_nh<<<nhBlk, 256, 0, stream>>>(menc, den);
    k_edge_max<<<edgeBlk, 256, 0, stream>>>(src, dst, als, ald, menc);
    k_decode_max<<<nhBlk, 256, 0, stream>>>(menc);
    k_edge_sum<<<edgeBlk, 256, 0, stream>>>(src, dst, als, ald, mF, den);
    k_edge_agg<<<aggBlk, 256, 0, stream>>>(src, dst, als, ald, mF, den, bufB, bufA);

    // ---- head ----
    k_final<<<(GN + 255) / 256, 256, 0, stream>>>(bufA, b2, fcw, fcb, out);
}